// EdgeDenseClassifierEdgeAttribute_9302899163220
// MI455X (gfx1250) — compile-verified
//
#include <hip/hip_runtime.h>
#include <hip/hip_bf16.h>

// D = A(16x4,f32) * B(4x16,f32) + C(16x16,f32) on gfx1250 wave32.
typedef __attribute__((ext_vector_type(2))) float v2f;
typedef __attribute__((ext_vector_type(8))) float v8f;

#define WAVES_PER_BLOCK 8
#define BLOCK_THREADS   256   // 8 wave32 waves

// Fast tanh: 1 - 2/(exp(2x)+1).  __expf -> v_exp_f32; rcp -> v_rcp_f32.
// Exact at the saturation limits (exp->inf => 1, exp->0 => -1), ~2ulp mid.
__device__ __forceinline__ float fast_tanh(float x) {
    float e = __expf(2.0f * x);
    return 1.0f - 2.0f * __builtin_amdgcn_rcpf(e + 1.0f);
}
__device__ __forceinline__ float fast_sigmoid(float z) {
    return __builtin_amdgcn_rcpf(1.0f + __expf(-z));
}

// One wave handles a tile of 16 edges.
//   H(16x16) = X(16x20) @ Wf(20x16)   via 5x V_WMMA_F32_16X16X4_F32
// where Wf[k][n] = (n<8) ? (k<16 ? W1[k][n]+W1[k+16][n] : W1[k+16][n]) : 0
// (the reference concatenates the SAME gathered row twice, so W1's two
//  embedding blocks fold into one 16x8 matrix; rows 16..19 of Wf are the
//  edge_attr block W1[32..35]).
__global__ __launch_bounds__(BLOCK_THREADS)
void edge_mlp_wmma_f32(const float* __restrict__ emb,    // [N_NODES,16]
                       const int*   __restrict__ eidx,   // [2,E], row 0 = src
                       const float* __restrict__ eattr,  // [E,4]
                       const float* __restrict__ W1,     // [36,8]
                       const float* __restrict__ b1,     // [8]
                       const float* __restrict__ W2,     // [8]
                       const float* __restrict__ b2,     // [1]
                       float* __restrict__ out,          // [E]
                       int E)
{
    // B operands pre-swizzled for the WMMA lane layout:
    //   step s, lane L (hi = L>=16, n = L&15):
    //     b.x = Wf[4s + 2*hi    ][n]
    //     b.y = Wf[4s + 2*hi + 1][n]
    __shared__ float Wlds[5][32][2];

    const int tid = threadIdx.x;
    if (tid < 5 * 32) {
        const int s    = tid >> 5;
        const int lane = tid & 31;
        const int hi   = lane >> 4;
        const int nn   = lane & 15;
        const int k0   = 4 * s + 2 * hi;
#pragma unroll
        for (int t = 0; t < 2; ++t) {
            const int k = k0 + t;            // 0..19
            float w = 0.0f;
            if (nn < 8) {
                w = W1[(k + 16) * 8 + nn];   // rows 16..35 of original W1
                if (k < 16) w += W1[k * 8 + nn];
            }
            Wlds[s][lane][t] = w;
        }
    }
    __syncthreads();

    const int lane = tid & 31;
    const int wave = tid >> 5;
    const int hi   = lane >> 4;   // 0: K pair {0,1}, 1: K pair {2,3}
    const int nn   = lane & 15;

    const int tile = blockIdx.x * WAVES_PER_BLOCK + wave;
    const int base = tile * 16;
    if (base >= E) return;        // wave-uniform: EXEC stays all-ones for WMMA

    // ---- B operands (LDS, conflict-free b64 per lane) ----
    v2f b[5];
#pragma unroll
    for (int s = 0; s < 5; ++s) {
        b[s].x = Wlds[s][lane][0];
        b[s].y = Wlds[s][lane][1];
    }

    // ---- A operands: this lane supplies row M = nn (edge base+nn),
    //      K pair {4s+2hi, 4s+2hi+1} for each of the 5 K-steps. ----
    int e_row = base + nn;
    if (e_row >= E) e_row = E - 1;               // clamp loads on ragged tail
    const long long node = (long long)eidx[e_row];
    const float* erow = emb + node * 16 + 2 * hi;

    v2f a[5];
#pragma unroll
    for (int s = 0; s < 4; ++s) {                // embedding features 0..15
        a[s].x = erow[4 * s + 0];                // one global_load_b64 each
        a[s].y = erow[4 * s + 1];
    }
    a[4].x = eattr[(long long)e_row * 4 + 2 * hi + 0];   // attr features 16..19
    a[4].y = eattr[(long long)e_row * 4 + 2 * hi + 1];

    // ---- 5x fp32 WMMA, K accumulation into C ----
    v8f acc = {};
#pragma unroll
    for (int s = 0; s < 5; ++s) {
        acc = __builtin_amdgcn_wmma_f32_16x16x4_f32(
            /*neg_a=*/false, a[s], /*neg_b=*/false, b[s],
            /*c_mod=*/(short)0, acc, /*reuse_a=*/false, /*reuse_b=*/false);
    }

    // ---- epilogue: h = tanh(acc + b1), scalar = h @ W2 + b2, sigmoid ----
    // acc[r] on lane (hi,nn) = H[edge base + r + 8*hi][nn]; columns nn>=8 are
    // the zero-padded ones.  Branch-free loads; mask only the contribution.
    const float b1n   = b1[nn & 7];
    const float w2n   = W2[nn & 7];
    const float msk   = (nn < 8) ? 1.0f : 0.0f;
    const float bias2 = b2[0];

    float S[8];
#pragma unroll
    for (int r = 0; r < 8; ++r) {
        float c = msk * (fast_tanh(acc[r] + b1n) * w2n);
        // reduce over the hidden dim (lane bits 0..3 inside each 16-lane half)
        c += __shfl_xor(c, 1, 32);
        c += __shfl_xor(c, 2, 32);
        c += __shfl_xor(c, 4, 32);
        c += __shfl_xor(c, 8, 32);
        S[r] = c;   // every lane of the half now holds all 8 edge sums
    }

    // lane n (n<8) of half hi emits edge base + n + 8*hi
    if (nn < 8) {
        float Sn = S[0];
#pragma unroll
        for (int r = 1; r < 8; ++r) Sn = (nn == r) ? S[r] : Sn;
        const int eo = base + nn + 8 * hi;
        if (eo < E) {
            out[eo] = fast_sigmoid(Sn + bias2);
        }
    }
}

extern "C" void kernel_launch(void* const* d_in, const int* in_sizes, int n_in,
                              void* d_out, int out_size, void* d_ws, size_t ws_size,
                              hipStream_t stream) {
    (void)n_in; (void)d_ws; (void)ws_size;
    const float* emb   = (const float*)d_in[0];
    const int*   eidx  = (const int*)  d_in[1];   // row 0 (first E entries) used
    const float* eattr = (const float*)d_in[2];
    const float* W1    = (const float*)d_in[3];
    const float* b1    = (const float*)d_in[4];
    const float* W2    = (const float*)d_in[5];
    const float* b2    = (const float*)d_in[6];
    float* out = (float*)d_out;

    const int E      = out_size;                       // one output per edge
    const int tiles  = (E + 15) / 16;                  // 16 edges per wave
    const int blocks = (tiles + WAVES_PER_BLOCK - 1) / WAVES_PER_BLOCK;
    (void)in_sizes;

    edge_mlp_wmma_f32<<<blocks, BLOCK_THREADS, 0, stream>>>(
        emb, eidx, eattr, W1, b1, W2, b2, out, E);
}